// DenseGenerator_75866302316779
// MI455X (gfx1250) — compile-verified
//
#include <hip/hip_runtime.h>

// Problem constants (fixed by the reference harness)
#define DD   32    // latent dim
#define NBAS 8     // num bases
#define BSZ  256   // batch
#define NS   128   // time steps
#define KT   32    // Taylor terms (||A||~0.45, t<1 -> error ~ 1e-40, << fp32 eps)

typedef __attribute__((ext_vector_type(2))) float v2f;
typedef __attribute__((ext_vector_type(8))) float v8f;

// One block per batch element. 128 threads = 4 wave32s.
//  phase 1: A = params @ W^T + bias           (all threads, fp32)
//  phase 2: M[k] = A^k z0 / k!, k=0..31       (wave 0, matvec chain via v_readlane)
//  phase 3: T[s][k] = t_s^k                   (all threads)
//  phase 4: out[s][o] = sum_k T[s][k]*M[k][o] via v_wmma_f32_16x16x4_f32
__global__ __launch_bounds__(128)
void expm_action_kernel(const float* __restrict__ z0,
                        const float* __restrict__ ts,
                        const float* __restrict__ kc,
                        const float* __restrict__ rc,
                        const float* __restrict__ W,
                        const float* __restrict__ bias,
                        float2* __restrict__ out)
{
    __shared__ float sP[2 * NBAS];
    __shared__ float sA[DD][DD + 1];     // padded: stride 33 avoids bank conflicts
    __shared__ float sM[KT][DD + 1];     // sM[k][o] = (A^k z0 / k!)[o]
    __shared__ float sT[NS][KT + 1];     // sT[s][k] = t_s^k

    const int tid = threadIdx.x;
    const int b   = blockIdx.x;

    // ---- phase 3: powers of t (one s per thread) ----
    {
        float t  = ts[tid];
        float pw = 1.0f;
        #pragma unroll
        for (int k = 0; k < KT; ++k) { sT[tid][k] = pw; pw *= t; }
    }
    // params -> LDS
    if (tid < 2 * NBAS)
        sP[tid] = (tid < NBAS) ? kc[b * NBAS + tid] : rc[b * NBAS + (tid - NBAS)];
    __syncthreads();

    // ---- phase 1: A[o][i] = bias[f] + sum_j p[j]*W[f*16+j], f = o*32+i ----
    #pragma unroll
    for (int e = 0; e < 8; ++e) {
        const int f   = tid * 8 + e;
        float     acc = bias[f];
        #pragma unroll
        for (int j = 0; j < 2 * NBAS; ++j) acc += sP[j] * W[f * (2 * NBAS) + j];
        sA[f >> 5][f & 31] = acc;
    }
    __syncthreads();

    // ---- phase 2: Taylor moment chain on wave 0 (lane o owns row o of A) ----
    if (tid < 32) {
        float row[DD];
        #pragma unroll
        for (int i = 0; i < DD; ++i) row[i] = sA[tid][i];
        float x = z0[b * DD + tid];      // m_0 = z0
        sM[0][tid] = x;
        #pragma unroll
        for (int k = 1; k < KT; ++k) {
            float y = 0.0f;
            #pragma unroll
            for (int i = 0; i < DD; ++i)
                y += row[i] * __shfl(x, i, 32);   // constant lane -> v_readlane
            x = y * (1.0f / (float)k);            // folded at compile time
            sM[k][tid] = x;
        }
    }
    __syncthreads();

    // ---- phase 4: per-wave 32x32 slab of the [128,32] = T[128,32] @ M[32,32] GEMM ----
    // ISA 7.12.2 fp32 layouts:
    //  A 16x4 : V0 = {lanes0-15: K=0 | lanes16-31: K=2}, V1 = {K=1 | K=3}, lane%16 = M row
    //  B 4x16 : V0 = {lanes0-15: K=0 | lanes16-31: K=2}, V1 = {K=1 | K=3}, lane%16 = N col
    //  C 16x16: VGPR r = {lanes0-15: M=r, N=lane | lanes16-31: M=r+8, N=lane-16}
    {
        const int w    = tid >> 5;       // wave id 0..3  -> rows 32w..32w+31
        const int lane = tid & 31;
        const int sub  = lane >> 4;      // half-wave selector
        const int m    = lane & 15;

        #pragma unroll
        for (int stile = 0; stile < 2; ++stile) {
            const int s0 = w * 32 + stile * 16;
            #pragma unroll
            for (int otile = 0; otile < 2; ++otile) {
                const int o0 = otile * 16;
                v8f c = {};
                #pragma unroll
                for (int kk = 0; kk < KT / 4; ++kk) {
                    const int k0 = kk * 4 + 2 * sub;
                    v2f a, bm;
                    a.x  = sT[s0 + m][k0];
                    a.y  = sT[s0 + m][k0 + 1];
                    bm.x = sM[k0][o0 + m];
                    bm.y = sM[k0 + 1][o0 + m];
                    // D = A x B + C   (8-arg form: neg_a, A, neg_b, B, c_mod, C, reuse_a, reuse_b)
                    c = __builtin_amdgcn_wmma_f32_16x16x4_f32(
                            false, a, false, bm, (short)0, c, false, false);
                }
                #pragma unroll
                for (int r = 0; r < 8; ++r) {
                    const int s   = s0 + r + 8 * sub;
                    const int o   = o0 + m;
                    const size_t idx = ((size_t)b * NS + s) * DD + o;
                    out[idx] = make_float2(c[r], 0.0f);   // complex64: (re, im=0)
                }
            }
        }
    }
}

extern "C" void kernel_launch(void* const* d_in, const int* in_sizes, int n_in,
                              void* d_out, int out_size, void* d_ws, size_t ws_size,
                              hipStream_t stream)
{
    (void)in_sizes; (void)n_in; (void)d_ws; (void)ws_size; (void)out_size;
    const float* z0   = (const float*)d_in[0];   // [B, D]
    const float* ts   = (const float*)d_in[1];   // [S]
    const float* kc   = (const float*)d_in[2];   // [B, NB]
    const float* rc   = (const float*)d_in[3];   // [B, NB]
    const float* W    = (const float*)d_in[4];   // [D*D, 2*NB]
    const float* bias = (const float*)d_in[5];   // [D*D]
    float2*      out  = (float2*)d_out;          // [B, S, D] complex64 -> (re, im) pairs

    expm_action_kernel<<<BSZ, 128, 0, stream>>>(z0, ts, kc, rc, W, bias, out);
}